// NeighborEmbedding_6408091206167
// MI455X (gfx1250) — compile-verified
//
#include <hip/hip_runtime.h>
#include <hip/hip_bf16.h>

// ---------------------------------------------------------------------------
// NeighborEmbedding on MI455X (gfx1250), fp32 via V_WMMA_F32_16X16X4_F32.
//
// Bandwidth roofline: d_0 (201MB) + d_1 (403MB) each read 3x (deg, d@xn,
// d^T@dx) => ~1.8GB => ~78us at 23.3 TB/s. FLOPs ~11.4G fp32 => WMMA is far
// from its ceiling; the design targets one clean streaming pass of d per
// GEMM. gemm1 double-buffers its 128x32 d-tiles through LDS with CDNA5
// GLOBAL_LOAD_ASYNC_TO_LDS_B128 (ASYNCcnt) to overlap the stream with WMMA.
// Deterministic: no float atomics; K-split partials + tree reduce.
// ---------------------------------------------------------------------------

typedef __attribute__((ext_vector_type(2))) float v2f;
typedef __attribute__((ext_vector_type(8))) float v8f;

#define WMMA_F32(a, b, c) \
  __builtin_amdgcn_wmma_f32_16x16x4_f32(false, (a), false, (b), (short)0, (c), false, false)

#if defined(__gfx1250__) && __has_builtin(__builtin_amdgcn_global_load_async_to_lds_b128) && \
    __has_builtin(__builtin_amdgcn_s_wait_asynccnt)
#define USE_ASYNC_LDS 1
typedef int v4i_ __attribute__((vector_size(4 * sizeof(int))));
typedef __attribute__((address_space(1))) v4i_ gv4i;   // global (AS1) int4
typedef __attribute__((address_space(3))) v4i_ lv4i;   // LDS (AS3) int4
#else
#define USE_ASYNC_LDS 0
#endif

#define Vn 4096
#define En 12288
#define Fn 8192
#define LDSP 36  // LDS row stride (floats): 16B-aligned rows + conflict-free A reads

// ---- stage 1: deg[n] = sum_m |d[m][n]| (2-stage, deterministic) -----------
__global__ void col_abssum_partial(const float* __restrict__ d, int N, int seg_rows,
                                   float* __restrict__ pdeg) {
  int n = blockIdx.x * blockDim.x + threadIdx.x;        // column
  int s = blockIdx.y;                                   // row segment
  size_t base = (size_t)s * seg_rows * N + n;
  float acc = 0.f;
  for (int m = 0; m < seg_rows; ++m) acc += fabsf(d[base + (size_t)m * N]);
  pdeg[(size_t)s * N + n] = acc;
}

__global__ void reduce_partials(const float* __restrict__ part, float* __restrict__ out,
                                int n, int S) {
  int i = blockIdx.x * blockDim.x + threadIdx.x;
  if (i >= n) return;
  float acc = 0.f;
  for (int s = 0; s < S; ++s) acc += part[(size_t)s * n + i];
  out[i] = acc;
}

// ---- xn = deg_inv * x ------------------------------------------------------
__global__ void scale_xn(const float* __restrict__ x, const float* __restrict__ deg,
                         float* __restrict__ xn) {
  int i = blockIdx.x * blockDim.x + threadIdx.x;
  int n = i >> 4;
  xn[i] = x[i] * rsqrtf(deg[n] + 1.0f);
}

// ---- GEMM1: dx = d @ xn  (M x K)·(K x 16), K-split partials ----------------
// 8 waves/block; each wave owns one 16-row tile. 128x32 d-tiles are double-
// buffered through LDS via async-to-LDS loads (ASYNCcnt) when available.
__global__ __launch_bounds__(256) void gemm1_dxn(const float* __restrict__ d,
                                                 const float* __restrict__ xn,
                                                 float* __restrict__ dxp,
                                                 int K, int seg_k) {
  __shared__ float tile[2][128 * LDSP];
  const int tid = threadIdx.x;
  const int lane = tid & 31, wv = tid >> 5;
  const int r = lane & 15, h = lane >> 4;
  const int mblk = blockIdx.x * 128;
  const int k0 = blockIdx.y * seg_k;
  const int nchunks = seg_k / 32;
  v8f acc = {};

#if USE_ASYNC_LDS
  // each thread moves 4 float4 groups per 128x32 tile (4096 floats / 256 thr)
  auto issue = [&](int buf, int kb) {
    #pragma unroll
    for (int j = 0; j < 4; ++j) {
      int g = tid + 256 * j;
      int row = g >> 3, c4 = (g & 7) * 4;
      __builtin_amdgcn_global_load_async_to_lds_b128(
          (gv4i*)(d + (size_t)(mblk + row) * K + kb + c4),
          (lv4i*)&tile[buf][row * LDSP + c4], 0, 0);
    }
  };
  issue(0, k0);
  for (int c = 0; c < nchunks; ++c) {
    const int buf = c & 1;
    if (c + 1 < nchunks) {
      issue(buf ^ 1, k0 + (c + 1) * 32);
      __builtin_amdgcn_s_wait_asynccnt(4);  // in-order => current tile landed
    } else {
      __builtin_amdgcn_s_wait_asynccnt(0);
    }
    __syncthreads();  // all waves' async writes for this tile visible
    const float* tw = &tile[buf][wv * 16 * LDSP];
    const int kb = k0 + c * 32;
    #pragma unroll
    for (int kk = 0; kk < 32; kk += 4) {
      v2f a, b;
      int ai = r * LDSP + kk + 2 * h;  // A: 16x4, VGPR0={K0,K2}, VGPR1={K1,K3}
      a.x = tw[ai];
      a.y = tw[ai + 1];
      const float* bp = &xn[(size_t)(kb + kk + 2 * h) * 16 + r];  // B: 4x16
      b.x = bp[0];
      b.y = bp[16];
      acc = WMMA_F32(a, b, acc);
    }
    __syncthreads();  // protect this buffer before iter c+1 re-issues into it
  }
#else
  for (int c = 0; c < nchunks; ++c) {
    const int kb = k0 + c * 32;
    __syncthreads();
    for (int l = tid; l < 128 * 8; l += 256) {  // coalesced float4 global -> LDS
      int row = l >> 3, c4 = (l & 7) * 4;
      *(float4*)&tile[0][row * LDSP + c4] =
          *(const float4*)&d[(size_t)(mblk + row) * K + kb + c4];
    }
    __syncthreads();
    const float* tw = &tile[0][wv * 16 * LDSP];
    #pragma unroll
    for (int kk = 0; kk < 32; kk += 4) {
      v2f a, b;
      int ai = r * LDSP + kk + 2 * h;
      a.x = tw[ai];
      a.y = tw[ai + 1];
      const float* bp = &xn[(size_t)(kb + kk + 2 * h) * 16 + r];
      b.x = bp[0];
      b.y = bp[16];
      acc = WMMA_F32(a, b, acc);
    }
  }
#endif

  float* outp = dxp + (size_t)blockIdx.y * ((size_t)gridDim.x * 128 * 16);
  const int mb = mblk + wv * 16;
  #pragma unroll
  for (int i = 0; i < 8; ++i)                        // C/D: VGPR i -> M=i+8h, N=r
    outp[(size_t)(mb + i + 8 * h) * 16 + r] = acc[i];
}

// ---- GEMM2: t = d^T @ dx  ((N x M-slice)·(M x 16)), K-split partials -------
// A-fragments read d row-major => naturally coalesced; no LDS needed.
__global__ __launch_bounds__(256) void gemm2_dTdx(const float* __restrict__ d,
                                                  const float* __restrict__ dx,
                                                  float* __restrict__ tp,
                                                  int N, int seg_m) {
  const int tid = threadIdx.x;
  const int lane = tid & 31, wv = tid >> 5;
  const int r = lane & 15, h = lane >> 4;
  const int ntile = blockIdx.x * 8 + wv;
  const int n = ntile * 16 + r;
  const int m0 = blockIdx.y * seg_m;
  v8f acc = {};
  #pragma unroll 4
  for (int m = m0; m < m0 + seg_m; m += 4) {
    v2f a, b;
    size_t ra = (size_t)(m + 2 * h) * N + n;        // A[n][k=m'] = d[m'][n]
    a.x = d[ra];
    a.y = d[ra + N];
    size_t rb = (size_t)(m + 2 * h) * 16 + r;       // B[k=m'][c] = dx[m'][c]
    b.x = dx[rb];
    b.y = dx[rb + 16];
    acc = WMMA_F32(a, b, acc);
  }
  float* outp = tp + (size_t)blockIdx.y * ((size_t)N * 16);
  #pragma unroll
  for (int i = 0; i < 8; ++i)
    outp[(size_t)(ntile * 16 + i + 8 * h) * 16 + r] = acc[i];
}

// ---- x_dual = x + deg_inv * (deg * xn - t) ---------------------------------
__global__ void finalize_dual(const float* __restrict__ x, const float* __restrict__ xn,
                              const float* __restrict__ t, const float* __restrict__ deg,
                              float* __restrict__ xd) {
  int i = blockIdx.x * blockDim.x + threadIdx.x;
  int n = i >> 4;
  float dg = deg[n];
  float dinv = rsqrtf(dg + 1.0f);
  xd[i] = x[i] + dinv * (dg * xn[i] - t[i]);
}

// ---- fused 3-layer MLP: 16 -> 128 (relu) -> 128 (relu) -> 128 --------------
// One block per 16 rows; wave w computes column tile [16w, 16w+16).
__global__ __launch_bounds__(256) void mlp3(const float* __restrict__ xin,
                                            const float* __restrict__ W1, const float* __restrict__ b1,
                                            const float* __restrict__ W2, const float* __restrict__ b2,
                                            const float* __restrict__ W3, const float* __restrict__ b3,
                                            float* __restrict__ out) {
  __shared__ float ha[16 * 132];
  __shared__ float hb[16 * 132];
  const int tid = threadIdx.x;
  const int lane = tid & 31, wv = tid >> 5;
  const int r = lane & 15, h = lane >> 4;
  const int cb = wv * 16;
  const int m0 = blockIdx.x * 16;

  // layer 1: K = 16, A straight from global (1KB tile, cache-resident)
  v8f a1 = {};
  #pragma unroll
  for (int k = 0; k < 16; k += 4) {
    v2f a, b;
    size_t ai = (size_t)(m0 + r) * 16 + k + 2 * h;
    a.x = xin[ai];
    a.y = xin[ai + 1];
    const float* wp = &W1[(size_t)(k + 2 * h) * 128 + cb + r];
    b.x = wp[0];
    b.y = wp[128];
    a1 = WMMA_F32(a, b, a1);
  }
  float bv1 = b1[cb + r];
  #pragma unroll
  for (int i = 0; i < 8; ++i) {
    float v = a1[i] + bv1;
    ha[(i + 8 * h) * 132 + cb + r] = v > 0.f ? v : 0.f;
  }
  __syncthreads();

  // layer 2: K = 128 from LDS
  v8f a2 = {};
  #pragma unroll 8
  for (int k = 0; k < 128; k += 4) {
    v2f a, b;
    int ai = r * 132 + k + 2 * h;
    a.x = ha[ai];
    a.y = ha[ai + 1];
    const float* wp = &W2[(size_t)(k + 2 * h) * 128 + cb + r];
    b.x = wp[0];
    b.y = wp[128];
    a2 = WMMA_F32(a, b, a2);
  }
  float bv2 = b2[cb + r];
  #pragma unroll
  for (int i = 0; i < 8; ++i) {
    float v = a2[i] + bv2;
    hb[(i + 8 * h) * 132 + cb + r] = v > 0.f ? v : 0.f;
  }
  __syncthreads();

  // layer 3: K = 128, no relu, write out
  v8f a3 = {};
  #pragma unroll 8
  for (int k = 0; k < 128; k += 4) {
    v2f a, b;
    int ai = r * 132 + k + 2 * h;
    a.x = hb[ai];
    a.y = hb[ai + 1];
    const float* wp = &W3[(size_t)(k + 2 * h) * 128 + cb + r];
    b.x = wp[0];
    b.y = wp[128];
    a3 = WMMA_F32(a, b, a3);
  }
  float bv3 = b3[cb + r];
  #pragma unroll
  for (int i = 0; i < 8; ++i)
    out[(size_t)(m0 + i + 8 * h) * 128 + cb + r] = a3[i] + bv3;
}

// ---------------------------------------------------------------------------
extern "C" void kernel_launch(void* const* d_in, const int* in_sizes, int n_in,
                              void* d_out, int out_size, void* d_ws, size_t ws_size,
                              hipStream_t stream) {
  const float* x_v = (const float*)d_in[0];
  const float* x_e = (const float*)d_in[1];
  const float* x_f = (const float*)d_in[2];
  const float* d0  = (const float*)d_in[3];  // (E, V)
  const float* d1  = (const float*)d_in[4];  // (F, E)
  // v_params / e_params / f_params flatten as (W1,b1,W2,b2,W3,b3) each
  const float* vp[6]; const float* ep[6]; const float* fp[6];
  for (int i = 0; i < 6; ++i) {
    vp[i] = (const float*)d_in[5 + i];
    ep[i] = (const float*)d_in[11 + i];
    fp[i] = (const float*)d_in[17 + i];
  }
  float* out_v = (float*)d_out;
  float* out_e = out_v + (size_t)Vn * 128;
  float* out_f = out_e + (size_t)En * 128;

  // workspace carve-up (floats); total ~8.5M floats (~34 MB)
  float* w = (float*)d_ws;
  float* deg0 = w;  w += Vn;
  float* deg1 = w;  w += En;
  float* pdeg0 = w; w += 64 * Vn;
  float* pdeg1 = w; w += 64 * En;
  float* xn0 = w;   w += Vn * 16;
  float* xn1 = w;   w += En * 16;
  float* dx0 = w;   w += En * 16;
  float* dx1 = w;   w += Fn * 16;
  float* t0 = w;    w += Vn * 16;
  float* t1 = w;    w += En * 16;
  float* xd0 = w;   w += Vn * 16;
  float* xd1 = w;   w += En * 16;
  float* pdx0 = w;  w += (size_t)8 * En * 16;
  float* pdx1 = w;  w += (size_t)16 * Fn * 16;
  float* pt0 = w;   w += (size_t)16 * Vn * 16;
  float* pt1 = w;   w += (size_t)8 * En * 16;

  // deg (column abs-sums), 2-stage
  col_abssum_partial<<<dim3(Vn / 256, 64), 256, 0, stream>>>(d0, Vn, En / 64, pdeg0);
  col_abssum_partial<<<dim3(En / 256, 64), 256, 0, stream>>>(d1, En, Fn / 64, pdeg1);
  reduce_partials<<<Vn / 256, 256, 0, stream>>>(pdeg0, deg0, Vn, 64);
  reduce_partials<<<En / 256, 256, 0, stream>>>(pdeg1, deg1, En, 64);

  // xn = deg_inv * x
  scale_xn<<<Vn * 16 / 256, 256, 0, stream>>>(x_v, deg0, xn0);
  scale_xn<<<En * 16 / 256, 256, 0, stream>>>(x_e, deg1, xn1);

  // dx0 = d0 @ xn0 (M=E, K=V, 8-way K-split) ; dx1 = d1 @ xn1 (M=F, K=E, 16-way)
  gemm1_dxn<<<dim3(En / 128, 8), 256, 0, stream>>>(d0, xn0, pdx0, Vn, Vn / 8);
  reduce_partials<<<En * 16 / 256, 256, 0, stream>>>(pdx0, dx0, En * 16, 8);
  gemm1_dxn<<<dim3(Fn / 128, 16), 256, 0, stream>>>(d1, xn1, pdx1, En, En / 16);
  reduce_partials<<<Fn * 16 / 256, 256, 0, stream>>>(pdx1, dx1, Fn * 16, 16);

  // t0 = d0^T @ dx0 (N=V, M=E, 16-way K-split) ; t1 = d1^T @ dx1 (N=E, M=F, 8-way)
  gemm2_dTdx<<<dim3(Vn / 128, 16), 256, 0, stream>>>(d0, dx0, pt0, Vn, En / 16);
  reduce_partials<<<Vn * 16 / 256, 256, 0, stream>>>(pt0, t0, Vn * 16, 16);
  gemm2_dTdx<<<dim3(En / 128, 8), 256, 0, stream>>>(d1, dx1, pt1, En, Fn / 8);
  reduce_partials<<<En * 16 / 256, 256, 0, stream>>>(pt1, t1, En * 16, 8);

  // x_dual
  finalize_dual<<<Vn * 16 / 256, 256, 0, stream>>>(x_v, xn0, t0, deg0, xd0);
  finalize_dual<<<En * 16 / 256, 256, 0, stream>>>(x_e, xn1, t1, deg1, xd1);

  // MLPs
  mlp3<<<Vn / 16, 256, 0, stream>>>(xd0, vp[0], vp[1], vp[2], vp[3], vp[4], vp[5], out_v);
  mlp3<<<En / 16, 256, 0, stream>>>(xd1, ep[0], ep[1], ep[2], ep[3], ep[4], ep[5], out_e);
  mlp3<<<Fn / 16, 256, 0, stream>>>(x_f, fp[0], fp[1], fp[2], fp[3], fp[4], fp[5], out_f);
}